// ToneMappingCurveLoss_16312285790284
// MI455X (gfx1250) — compile-verified
//
#include <hip/hip_runtime.h>
#include <math.h>

// Problem geometry: [B=16, C=3, H=512, W=512] fp32
#define HWSZ    262144            // 512*512 (2^18)
#define NPIX    4194304           // 16*512*512 luma pixels
#define TILE    512               // pixels per workgroup tile
#define NTILES  (NPIX / TILE)     // 8192
#define THREADS 256               // 8 wave32s
#define PT      2                 // pixels per thread per tile (b64 = 2 floats)
#define NBINS   16
#define NWAVES  (THREADS / 32)
#define GRID    2048              // each block handles 4 tiles, double buffered

// ---- CDNA5 async global->LDS path -----------------------------------------

// Low 32 bits of a generic LDS pointer == byte offset within the wave's LDS
// allocation (ISA: LDS aperture).
__device__ __forceinline__ unsigned lds_addr32(const void* p) {
  return (unsigned)(unsigned long long)p;
}

// global_load_async_to_lds_b64: VDST = LDS byte address VGPR,
// VADDR = 32-bit unsigned byte offset, SADDR = 64-bit base (GVS mode).
// Tracked by ASYNCcnt; loads complete (decrement) in issue order per wave.
__device__ __forceinline__ void async_copy_b64(unsigned lds_off, unsigned voff,
                                               const float* sbase) {
  unsigned long long s = (unsigned long long)sbase;
  asm volatile("global_load_async_to_lds_b64 %0, %1, %2 offset:0"
               :
               : "v"(lds_off), "v"(voff), "s"(s)
               : "memory");
}

// asm-only waits: the "memory" clobber is essential — it stops clang from
// hoisting the tile ds_loads above the hardware wait.
__device__ __forceinline__ void wait_async_le9() {
  asm volatile("s_wait_asynccnt 9" ::: "memory");
}
__device__ __forceinline__ void wait_async_le0() {
  asm volatile("s_wait_asynccnt 0" ::: "memory");
}

// ---- kernels ---------------------------------------------------------------

__global__ void tml_init_kernel(float* __restrict__ ws) {
  int i = threadIdx.x;
  if (i < 3 * NBINS) ws[i] = 0.0f;
}

__global__ __launch_bounds__(THREADS) void tml_hist_kernel(
    const float* __restrict__ pred, const float* __restrict__ targ,
    const float* __restrict__ img, float* __restrict__ ws) {
  // Double-buffered streaming tile: [buf][src*3+chan][pixel] = 36 KB
  __shared__ float buf[2][9][TILE];
  // Per-wave private histograms: no inter-wave LDS atomic contention.
  __shared__ float hist[NWAVES][3][NBINS];

  const int tid  = threadIdx.x;
  const int wave = tid >> 5;

  for (int i = tid; i < NWAVES * 3 * NBINS; i += THREADS)
    (&hist[0][0][0])[i] = 0.0f;
  __syncthreads();

  const float* bases[3] = {pred, targ, img};
  const unsigned voff = (unsigned)(tid * PT * sizeof(float));  // bytes

  // Issue 9 async b64 copies for tile t into buffer bb. Each thread loads
  // exactly the LDS bytes it will later read itself -> ASYNCcnt is a valid
  // per-wave sync, no workgroup barrier needed in the streaming loop.
  auto issue = [&](int t, int bb) {
    const int p0   = t * TILE;
    const int im   = p0 >> 18;          // / HWSZ  (tiles never straddle planes)
    const int pix0 = p0 & (HWSZ - 1);   // % HWSZ
#pragma unroll
    for (int s = 0; s < 3; ++s) {
#pragma unroll
      for (int c = 0; c < 3; ++c) {
        const float* gp = bases[s] + (size_t)(im * 3 + c) * HWSZ + pix0;
        const unsigned loff = lds_addr32(&buf[bb][s * 3 + c][0]) + voff;
        async_copy_b64(loff, voff, gp);
      }
    }
  };

  auto process = [&](int bb) {
#pragma unroll
    for (int j = 0; j < PT; ++j) {
      const int idx = tid * PT + j;
      const float pl = 0.299f * buf[bb][0][idx] + 0.587f * buf[bb][1][idx] +
                       0.114f * buf[bb][2][idx];
      const float tl = 0.299f * buf[bb][3][idx] + 0.587f * buf[bb][4][idx] +
                       0.114f * buf[bb][5][idx];
      const float il = 0.299f * buf[bb][6][idx] + 0.587f * buf[bb][7][idx] +
                       0.114f * buf[bb][8][idx];
      if (il < 1.0f) {  // il >= 1.0 falls in no bin (discarded segment)
        int bin = (int)floorf(il * (float)NBINS);
        bin = bin < 0 ? 0 : (bin > NBINS - 1 ? NBINS - 1 : bin);
        atomicAdd(&hist[wave][0][bin], 1.0f);  // ds_add_f32
        atomicAdd(&hist[wave][1][bin], pl);
        atomicAdd(&hist[wave][2][bin], tl);
      }
    }
  };

  int t  = (int)blockIdx.x;   // GRID <= NTILES, so always valid
  int bb = 0;
  issue(t, 0);
  for (; t < NTILES; t += (int)gridDim.x) {
    const int tn = t + (int)gridDim.x;
    if (tn < NTILES) {
      issue(tn, bb ^ 1);   // 9 new in flight (<= 18 total)
      wait_async_le9();    // oldest 9 (tile t) complete, in order
    } else {
      wait_async_le0();
    }
    process(bb);
    bb ^= 1;
  }

  __syncthreads();
  // Cross-wave reduce + one global atomic per {quantity,bin} per block.
  for (int i = tid; i < 3 * NBINS; i += THREADS) {
    const int s = i >> 4;
    const int b = i & (NBINS - 1);
    float acc = 0.0f;
#pragma unroll
    for (int w = 0; w < NWAVES; ++w) acc += hist[w][s][b];
    atomicAdd(&ws[i], acc);
  }
}

__global__ void tml_final_kernel(const float* __restrict__ ws,
                                 float* __restrict__ out) {
  if (threadIdx.x == 0) {
    float s = 0.0f;
    for (int b = 0; b < NBINS; ++b) {
      const float c  = ws[b];
      const float p  = ws[NBINS + b];
      const float t2 = ws[2 * NBINS + b];
      const float safe = fmaxf(c, 1.0f);
      s += (c > 0.0f) ? fabsf(p / safe - t2 / safe) : 0.0f;
    }
    out[0] = s / (float)NBINS;
  }
}

// ---- launch ----------------------------------------------------------------

extern "C" void kernel_launch(void* const* d_in, const int* in_sizes, int n_in,
                              void* d_out, int out_size, void* d_ws,
                              size_t ws_size, hipStream_t stream) {
  const float* pred = (const float*)d_in[0];
  const float* targ = (const float*)d_in[1];
  const float* img  = (const float*)d_in[2];
  float* out = (float*)d_out;
  float* ws  = (float*)d_ws;

  tml_init_kernel<<<1, 64, 0, stream>>>(ws);
  tml_hist_kernel<<<GRID, THREADS, 0, stream>>>(pred, targ, img, ws);
  tml_final_kernel<<<1, 32, 0, stream>>>(ws, out);
}